// TrainableSinkhornSort_28080496181698
// MI455X (gfx1250) — compile-verified
//
#include <hip/hip_runtime.h>

#define N_ITERS 5
#define EPSF 1e-8f
#define NMAT 128
#define LDSTRIDE 132   // padded row stride (floats): distinct banks for strided A-fragment loads

typedef __attribute__((ext_vector_type(2))) float v2f;
typedef __attribute__((ext_vector_type(8))) float v8f;

// One workgroup (256 threads = 8 wave32) per 128x128 tile. Whole tile lives in LDS.
// Row/col reductions use V_WMMA_F32_16X16X4_F32 with data in A and ones in B.
__launch_bounds__(256, 2)
__global__ void sinkhorn_kernel(const float* __restrict__ scores,
                                const float* __restrict__ tau_p,
                                float* __restrict__ out)
{
    __shared__ float P[NMAT * LDSTRIDE];
    __shared__ float invrow[NMAT];
    __shared__ float invcol[NMAT];

    const int b    = blockIdx.x;
    const int t    = threadIdx.x;
    const int wave = t >> 5;          // 0..7 : slab (rows) / stripe (cols) owner
    const int lane = t & 31;
    const int half = lane >> 4;       // K-half select for A fragment (K=0,1 vs K=2,3)
    const int m    = lane & 15;       // M index within 16

    // exp(x/tau) = exp2(x * log2(e)/tau)  -> v_exp_f32
    const float c2 = (1.0f / tau_p[0]) * 1.44269504088896340736f;

    // ---- Load tile from HBM (read exactly once), fuse exp, park in LDS ----
    const float4* gsrc = (const float4*)(scores + (size_t)b * NMAT * NMAT);
    #pragma unroll 4
    for (int sweep = 0; sweep < 16; ++sweep) {
        int g = sweep * 1024 + t * 4;          // flat float index, coalesced b128 loads
        float4 v = gsrc[g >> 2];
        int row = g >> 7, col = g & 127;
        float4 e;
        e.x = exp2f(v.x * c2); e.y = exp2f(v.y * c2);
        e.z = exp2f(v.z * c2); e.w = exp2f(v.w * c2);
        *(float4*)&P[row * LDSTRIDE + col] = e;
    }

    const v2f onesB = {1.0f, 1.0f};            // B = ones(4x16): layout-independent
    const int base16 = wave * 16;

    for (int it = 0; it < N_ITERS; ++it) {
        __syncthreads();

        // ---- row sums: D = A(P slab 16x4) x ones ; wave owns rows [base16, base16+16) ----
        {
            v8f acc = {};
            for (int kc = 0; kc < 32; ++kc) {
                int kb = kc * 4 + half * 2;
                const float* src = &P[(base16 + m) * LDSTRIDE + kb];
                v2f a; a.x = src[0]; a.y = src[1];          // A[m][kb], A[m][kb+1]
                acc = __builtin_amdgcn_wmma_f32_16x16x4_f32(
                        false, a, false, onesB, (short)0, acc, false, false);
            }
            if (m == 0) {                                   // lanes 0 & 16: M = v (+8*half)
                int base = base16 + half * 8;
                #pragma unroll
                for (int v = 0; v < 8; ++v)
                    invrow[base + v] = 1.0f / (acc[v] + EPSF);
            }
        }
        __syncthreads();

        // ---- row normalize: thread owns half a row (64 contiguous floats) ----
        {
            int row = t >> 1, cb = (t & 1) * 64;
            float inv = invrow[row];
            float4* prow = (float4*)&P[row * LDSTRIDE + cb];
            #pragma unroll
            for (int k = 0; k < 16; ++k) {
                float4 v = prow[k];
                v.x *= inv; v.y *= inv; v.z *= inv; v.w *= inv;
                prow[k] = v;
            }
        }
        __syncthreads();

        // ---- col sums = row sums of P^T: A[mm][kk] = P[kc*4+kk][stripe+mm] ----
        {
            v8f acc = {};
            for (int kc = 0; kc < 32; ++kc) {
                int kb = kc * 4 + half * 2;
                const float* src = &P[kb * LDSTRIDE + base16 + m];
                v2f a; a.x = src[0]; a.y = src[LDSTRIDE];
                acc = __builtin_amdgcn_wmma_f32_16x16x4_f32(
                        false, a, false, onesB, (short)0, acc, false, false);
            }
            if (m == 0) {
                int base = base16 + half * 8;
                #pragma unroll
                for (int v = 0; v < 8; ++v) {
                    float S  = acc[v];
                    float ic = 1.0f / (S + EPSF);
                    invcol[base + v] = ic;
                    // colsum after the final col-normalize is S/(S+eps): emit directly,
                    // eliding the last normalize pass and a 6th reduction pass.
                    if (it == N_ITERS - 1)
                        out[(size_t)b * NMAT + base + v] = S * ic;
                }
            }
        }
        if (it == N_ITERS - 1) break;
        __syncthreads();

        // ---- col normalize ----
        {
            int row = t >> 1, cb = (t & 1) * 64;
            float4* prow = (float4*)&P[row * LDSTRIDE + cb];
            const float* ic = &invcol[cb];
            #pragma unroll
            for (int k = 0; k < 16; ++k) {
                float4 v = prow[k];
                v.x *= ic[k * 4 + 0]; v.y *= ic[k * 4 + 1];
                v.z *= ic[k * 4 + 2]; v.w *= ic[k * 4 + 3];
                prow[k] = v;
            }
        }
    }
}

extern "C" void kernel_launch(void* const* d_in, const int* in_sizes, int n_in,
                              void* d_out, int out_size, void* d_ws, size_t ws_size,
                              hipStream_t stream) {
    const float* scores = (const float*)d_in[0];
    const float* tau    = (const float*)d_in[1];
    float* out          = (float*)d_out;
    int batches = in_sizes[0] / (NMAT * NMAT);   // 4096
    sinkhorn_kernel<<<dim3(batches), dim3(256), 0, stream>>>(scores, tau, out);
}